// NeuralMirrorModule_19988777796233
// MI455X (gfx1250) — compile-verified
//
#include <hip/hip_runtime.h>

// ---------------------------------------------------------------------------
// Bregman divergence D_h(y || y0), elementwise over N = 4*1024*512 floats,
// with a 126-neuron inner sum (6 activation groups x 21 neurons).
// Transcendental-bound kernel: ~210 TRANS ops/element. No GEMM structure ->
// WMMA inapplicable; optimize the TRANS/VALU path instead.
//   - per-neuron constants precomputed once (division-free hot loop),
//     fetched via wave-uniform s_load (SALU/SMEM path, parallel to VALU)
//   - degenerate-w fallback behind a wave-uniform scalar branch (zero cost
//     on the common path)
//   - raw v_sqrt_f32 / v_log_f32 / v_exp_f32 via fast builtins
// ---------------------------------------------------------------------------

constexpr int   NJ        = 126;
constexpr int   NGRP      = 21;
constexpr float EPS       = 1e-3f;
constexpr float EPS_PROB  = 1e-10f;
constexpr float ONE_THIRD = 1.0f / 3.0f;

#if __has_builtin(__builtin_amdgcn_exp2f)
#define FAST_EXP2(x) __builtin_amdgcn_exp2f(x)
#else
#define FAST_EXP2(x) exp2f(x)
#endif

#if __has_builtin(__builtin_amdgcn_sqrtf)
#define FAST_SQRT(x) __builtin_amdgcn_sqrtf(x)
#else
#define FAST_SQRT(x) sqrtf(x)
#endif

// Per-neuron precomputed constants, 8 floats per neuron in d_ws:
//   [0]=w  [1]=b  [2]=kS  [3]=v  [4]=small_flag  [5..7]=pad
//   g0: kS = v*w/4            (factored quartic primitive, division-free)
//   g1: kS = v*w/3            (factored cubic primitive, division-free)
//   g2: kS = small ? v : v*(2/3)/w
//   g3: kS = small ? v : v*(3/4)/w
//   g4: kS = small ? v : v/w
//   g5: kS = small ? v : v/w
__global__ void nm_init_tbl(const float* __restrict__ v,
                            const float* __restrict__ w,
                            const float* __restrict__ b,
                            float* __restrict__ tbl) {
    int j = blockIdx.x * blockDim.x + threadIdx.x;
    if (j >= NJ) return;
    int g = j / NGRP;
    float wj = w[j], bj = b[j], vj = v[j];
    bool  small = fabsf(wj) < 1e-12f;
    float rw = small ? 1.0f : (1.0f / wj);
    float kS;
    if (g == 0)      kS = vj * wj * 0.25f;
    else if (g == 1) kS = vj * wj * (1.0f / 3.0f);
    else if (g == 2) kS = small ? vj : vj * (2.0f / 3.0f) * rw;
    else if (g == 3) kS = small ? vj : vj * 0.75f * rw;
    else             kS = small ? vj : vj * rw;   // g4, g5
    float* t = tbl + j * 8;
    t[0] = wj; t[1] = bj; t[2] = kS; t[3] = vj;
    t[4] = small ? 1.0f : 0.0f;
    t[5] = 0.0f; t[6] = 0.0f; t[7] = 0.0f;
}

__global__ __launch_bounds__(256) void nm_bregman(
    const float* __restrict__ y,  const float* __restrict__ y0,
    const float* __restrict__ tbl,
    const float* __restrict__ pa, const float* __restrict__ pc,
    float* __restrict__ out, int n) {
    int i    = blockIdx.x * blockDim.x + threadIdx.x;
    int base = i * 4;
    if (base >= n) return;

    // CDNA5 prefetch path (global_prefetch_b8): warm the streaming lines.
    __builtin_prefetch(y  + base, 0, 0);
    __builtin_prefetch(y0 + base, 0, 0);

    float ye[4], z0[4];
    bool full = (base + 3) < n;
    if (full) {
        float4 a4 = *reinterpret_cast<const float4*>(y  + base);
        float4 b4 = *reinterpret_cast<const float4*>(y0 + base);
        ye[0] = a4.x; ye[1] = a4.y; ye[2] = a4.z; ye[3] = a4.w;
        z0[0] = b4.x; z0[1] = b4.y; z0[2] = b4.z; z0[3] = b4.w;
    } else {
#pragma unroll
        for (int k = 0; k < 4; ++k) {
            int idx = base + k;
            bool ok = idx < n;
            ye[k] = ok ? y[idx]  : 0.5f;
            z0[k] = ok ? y0[idx] : 0.5f;
        }
    }

    float dy[4], dy2[4], acc[4];
#pragma unroll
    for (int k = 0; k < 4; ++k) {
        dy[k]  = ye[k] - z0[k];
        dy2[k] = dy[k] * dy[k];
        acc[k] = 0.0f;
    }

    // ---- group 0: act = u^3. Exact factorization (for w != small):
    //   (u^4 - u0^4)/(4w) - u0^3*dy  ==  (v*w/4) * dy^2 * (u^2 + 2*u*u0 + 3*u0^2)
    for (int j = 0; j < NGRP; ++j) {
        const float* t = tbl + j * 8;
        float w = t[0], b = t[1], kS = t[2];
#pragma unroll
        for (int k = 0; k < 4; ++k) {
            float u  = fmaf(w, ye[k], b);
            float u0 = fmaf(w, z0[k], b);
            float poly = fmaf(u, fmaf(2.0f, u0, u), 3.0f * u0 * u0);
            acc[k] = fmaf(kS * dy2[k], poly, acc[k]);
        }
    }

    // ---- group 1: act = u^2.  term == (v*w/3) * dy^2 * (u + 2*u0)
    for (int j = NGRP; j < 2 * NGRP; ++j) {
        const float* t = tbl + j * 8;
        float w = t[0], b = t[1], kS = t[2];
#pragma unroll
        for (int k = 0; k < 4; ++k) {
            float u  = fmaf(w, ye[k], b);
            float u0 = fmaf(w, z0[k], b);
            acc[k] = fmaf(kS * dy2[k], fmaf(2.0f, u0, u), acc[k]);
        }
    }

    // ---- group 2: act = sqrt(max(u,0)); prim = (2/3) m^1.5 / w = (2/3) m*s / w
    for (int j = 2 * NGRP; j < 3 * NGRP; ++j) {
        const float* t = tbl + j * 8;
        float w = t[0], b = t[1], kS = t[2], kV = t[3];
        if (__builtin_expect(t[4] == 0.0f, 1)) {   // wave-uniform branch
#pragma unroll
            for (int k = 0; k < 4; ++k) {
                float u  = fmaf(w, ye[k], b);
                float u0 = fmaf(w, z0[k], b);
                float m  = fmaxf(u, 0.0f),  m0 = fmaxf(u0, 0.0f);
                float s  = FAST_SQRT(m),    s0 = FAST_SQRT(m0);
                acc[k] = fmaf(kS, m * s - m0 * s0,
                              fmaf(-kV, s0 * dy[k], acc[k]));
            }
        } else {                                   // |w| < 1e-12 fallback
#pragma unroll
            for (int k = 0; k < 4; ++k) {
                float u  = fmaf(w, ye[k], b);
                float u0 = fmaf(w, z0[k], b);
                float s  = FAST_SQRT(fmaxf(u, 0.0f));
                float s0 = FAST_SQRT(fmaxf(u0, 0.0f));
                acc[k] = fmaf(kS, s * ye[k] - s0 * z0[k],
                              fmaf(-kV, s0 * dy[k], acc[k]));
            }
        }
    }

    // ---- group 3: act = cbrt(max(u,0)) = exp2(log2(m)/3); prim = 0.75 m^(4/3)/w
    for (int j = 3 * NGRP; j < 4 * NGRP; ++j) {
        const float* t = tbl + j * 8;
        float w = t[0], b = t[1], kS = t[2], kV = t[3];
        if (__builtin_expect(t[4] == 0.0f, 1)) {
#pragma unroll
            for (int k = 0; k < 4; ++k) {
                float u  = fmaf(w, ye[k], b);
                float u0 = fmaf(w, z0[k], b);
                float m   = fmaxf(u, 0.0f),  m0 = fmaxf(u0, 0.0f);
                float cb  = FAST_EXP2(ONE_THIRD * __log2f(m));   // cbrt; m=0 -> 0
                float cb0 = FAST_EXP2(ONE_THIRD * __log2f(m0));
                acc[k] = fmaf(kS, m * cb - m0 * cb0,             // m^(4/3)
                              fmaf(-kV, cb0 * dy[k], acc[k]));
            }
        } else {
#pragma unroll
            for (int k = 0; k < 4; ++k) {
                float u  = fmaf(w, ye[k], b);
                float u0 = fmaf(w, z0[k], b);
                float cb  = FAST_EXP2(ONE_THIRD * __log2f(fmaxf(u, 0.0f)));
                float cb0 = FAST_EXP2(ONE_THIRD * __log2f(fmaxf(u0, 0.0f)));
                acc[k] = fmaf(kS, cb * ye[k] - cb0 * z0[k],
                              fmaf(-kV, cb0 * dy[k], acc[k]));
            }
        }
    }

    // ---- group 4: act = log(max(u,0)+eps); prim = (us*log us - us)/w
    for (int j = 4 * NGRP; j < 5 * NGRP; ++j) {
        const float* t = tbl + j * 8;
        float w = t[0], b = t[1], kS = t[2], kV = t[3];
        if (__builtin_expect(t[4] == 0.0f, 1)) {
#pragma unroll
            for (int k = 0; k < 4; ++k) {
                float u  = fmaf(w, ye[k], b);
                float u0 = fmaf(w, z0[k], b);
                float us  = fmaxf(u, 0.0f)  + EPS;
                float us0 = fmaxf(u0, 0.0f) + EPS;
                float l  = __logf(us),  l0 = __logf(us0);
                acc[k] = fmaf(kS, us * (l - 1.0f) - us0 * (l0 - 1.0f),
                              fmaf(-kV, l0 * dy[k], acc[k]));
            }
        } else {
#pragma unroll
            for (int k = 0; k < 4; ++k) {
                float u  = fmaf(w, ye[k], b);
                float u0 = fmaf(w, z0[k], b);
                float l  = __logf(fmaxf(u, 0.0f)  + EPS);
                float l0 = __logf(fmaxf(u0, 0.0f) + EPS);
                acc[k] = fmaf(kS, l * ye[k] - l0 * z0[k],
                              fmaf(-kV, l0 * dy[k], acc[k]));
            }
        }
    }

    // ---- group 5: act = exp(u); prim = exp(u)/w
    for (int j = 5 * NGRP; j < 6 * NGRP; ++j) {
        const float* t = tbl + j * 8;
        float w = t[0], b = t[1], kS = t[2], kV = t[3];
        if (__builtin_expect(t[4] == 0.0f, 1)) {
#pragma unroll
            for (int k = 0; k < 4; ++k) {
                float u  = fmaf(w, ye[k], b);
                float u0 = fmaf(w, z0[k], b);
                float e  = __expf(u), e0 = __expf(u0);
                acc[k] = fmaf(kS, e - e0, fmaf(-kV, e0 * dy[k], acc[k]));
            }
        } else {
#pragma unroll
            for (int k = 0; k < 4; ++k) {
                float u  = fmaf(w, ye[k], b);
                float u0 = fmaf(w, z0[k], b);
                float e  = __expf(u), e0 = __expf(u0);
                acc[k] = fmaf(kS, e * ye[k] - e0 * z0[k],
                              fmaf(-kV, e0 * dy[k], acc[k]));
            }
        }
    }

    // ---- scalar a/c terms: 0.5*a*dy^2 + c*(y*(log ys - log y0s) - dy)
    float ha = 0.5f * pa[0];
    float cc = pc[0];
#pragma unroll
    for (int k = 0; k < 4; ++k) {
        acc[k] = fmaf(ha, dy2[k], acc[k]);
        float ls  = __logf(fmaxf(ye[k], EPS_PROB));
        float l0s = __logf(fmaxf(z0[k], EPS_PROB));
        acc[k] = fmaf(cc, fmaf(ye[k], ls - l0s, -dy[k]), acc[k]);
    }

    if (full) {
        *reinterpret_cast<float4*>(out + base) =
            make_float4(acc[0], acc[1], acc[2], acc[3]);
    } else {
#pragma unroll
        for (int k = 0; k < 4; ++k)
            if (base + k < n) out[base + k] = acc[k];
    }
}

extern "C" void kernel_launch(void* const* d_in, const int* in_sizes, int n_in,
                              void* d_out, int out_size, void* d_ws, size_t ws_size,
                              hipStream_t stream) {
    // setup_inputs() order: y, y0, v, w, b, a, c
    const float* y  = (const float*)d_in[0];
    const float* y0 = (const float*)d_in[1];
    const float* v  = (const float*)d_in[2];
    const float* w  = (const float*)d_in[3];
    const float* b  = (const float*)d_in[4];
    const float* a  = (const float*)d_in[5];
    const float* c  = (const float*)d_in[6];
    float* out = (float*)d_out;
    float* tbl = (float*)d_ws;              // 126*8*4 = 4032 bytes of scratch

    int n = in_sizes[0];                    // 4*1024*512 = 2,097,152

    nm_init_tbl<<<1, 128, 0, stream>>>(v, w, b, tbl);

    int n4      = (n + 3) / 4;              // elements per thread = 4
    int threads = 256;
    int blocks  = (n4 + threads - 1) / threads;
    nm_bregman<<<blocks, threads, 0, stream>>>(y, y0, tbl, a, c, out, n);
}